// SchattenSimilarityLoss_65025804861946
// MI455X (gfx1250) — compile-verified
//
#include <hip/hip_runtime.h>
#include <math.h>

// Schatten-2 similarity loss via Gram-matrix trace identity:
//   ||A B^T||_F^2 = <A^T A, B^T B>_F
// Per batch only 3 scalars are needed: Saa=||G_A||_F^2, Sbb=||G_B||_F^2,
// Sab=<G_A,G_B>. G (384x384) is processed in 32x32 blocks (= 2x2 WMMA tiles,
// register-blocked so each loaded fragment feeds two v_wmma_f32_16x16x4_f32).
// G is symmetric: full sum = diag blocks + 2 * upper off-diag blocks.
// Block tiles live entirely in accumulator VGPRs, are squared/dotted in
// registers, wave-reduced, and atomically accumulated per batch.

typedef __attribute__((ext_vector_type(2))) float v2f;
typedef __attribute__((ext_vector_type(8))) float v8f;

#define NBATCH 128
#define SEQ    256
#define DIM    384
#define TB     12            // DIM / 32 (32-wide Gram blocks)
#define NPAIRS 78            // TB*(TB+1)/2
#define BLOCK  256           // 8 wave32 per block
#define WPB    (BLOCK / 32)

#define WMMA_F32(A, B, C) \
    __builtin_amdgcn_wmma_f32_16x16x4_f32(false, (A), false, (B), (short)0, (C), false, false)

__global__ void schatten_zero_acc(float* __restrict__ acc) {
    int i = threadIdx.x;
    if (i < NBATCH * 3) acc[i] = 0.0f;
}

__global__ __launch_bounds__(BLOCK)
void schatten_gram_tiles(const float* __restrict__ A,
                         const float* __restrict__ Bm,
                         float* __restrict__ acc) {
    const int wave  = blockIdx.x * WPB + (threadIdx.x >> 5);
    const int lane  = threadIdx.x & 31;
    const int batch = wave / NPAIRS;
    if (batch >= NBATCH) return;          // wave-uniform guard (EXEC stays all-1)
    int p = wave - batch * NPAIRS;

    // decode upper-triangle block index p -> (bi, bj), bi <= bj
    int bi = 0;
    while (p >= TB - bi) { p -= TB - bi; ++bi; }
    const int bj = bi + p;

    const float* __restrict__ Ab = A  + (size_t)batch * SEQ * DIM;
    const float* __restrict__ Bb = Bm + (size_t)batch * SEQ * DIM;

    // 32-bit 16x4 A-fragment layout: lanes 0-15 carry K=0,1; lanes 16-31 K=2,3;
    // lane-within-half indexes M (or N for the B operand). The identical load
    // pattern serves as both WMMA operands, only the column base differs.
    const int col    = lane & 15;
    const int rowoff = (lane >> 4) << 1;
    const int di0    = bi * 32 + col;
    const int di1    = di0 + 16;
    const int dj0    = bj * 32 + col;
    const int dj1    = dj0 + 16;

    // 2x2 tile accumulators for the 32x32 block of G_A and of G_B
    v8f ga00 = {}, ga01 = {}, ga10 = {}, ga11 = {};
    v8f gb00 = {}, gb01 = {}, gb10 = {}, gb11 = {};

    for (int k0 = 0; k0 < SEQ; k0 += 4) {
        const float* ra0 = Ab + (size_t)(k0 + rowoff) * DIM;
        const float* ra1 = ra0 + DIM;
        const float* rb0 = Bb + (size_t)(k0 + rowoff) * DIM;
        const float* rb1 = rb0 + DIM;

        // prefetch next K-chunk (speculative; lowers to global_prefetch_b8)
        __builtin_prefetch(ra0 + 4 * DIM + di0, 0, 0);
        __builtin_prefetch(rb0 + 4 * DIM + di0, 0, 0);

        v2f ai0 = { ra0[di0], ra1[di0] };
        v2f ai1 = { ra0[di1], ra1[di1] };
        v2f aj0 = { ra0[dj0], ra1[dj0] };
        v2f aj1 = { ra0[dj1], ra1[dj1] };
        v2f bi0 = { rb0[di0], rb1[di0] };
        v2f bi1 = { rb0[di1], rb1[di1] };
        v2f bj0 = { rb0[dj0], rb1[dj0] };
        v2f bj1 = { rb0[dj1], rb1[dj1] };

        ga00 = WMMA_F32(ai0, aj0, ga00);
        ga01 = WMMA_F32(ai0, aj1, ga01);
        ga10 = WMMA_F32(ai1, aj0, ga10);
        ga11 = WMMA_F32(ai1, aj1, ga11);
        gb00 = WMMA_F32(bi0, bj0, gb00);
        gb01 = WMMA_F32(bi0, bj1, gb01);
        gb10 = WMMA_F32(bi1, bj0, gb10);
        gb11 = WMMA_F32(bi1, bj1, gb11);
    }

    // per-lane partial Frobenius inner products over all 4 tiles x 8 rows
    float pAA = 0.0f, pBB = 0.0f, pAB = 0.0f;
#pragma unroll
    for (int i = 0; i < 8; ++i) {
        pAA += ga00[i] * ga00[i] + ga01[i] * ga01[i]
             + ga10[i] * ga10[i] + ga11[i] * ga11[i];
        pBB += gb00[i] * gb00[i] + gb01[i] * gb01[i]
             + gb10[i] * gb10[i] + gb11[i] * gb11[i];
        pAB += ga00[i] * gb00[i] + ga01[i] * gb01[i]
             + ga10[i] * gb10[i] + ga11[i] * gb11[i];
    }
    const float w = (bi == bj) ? 1.0f : 2.0f;   // block-symmetry weight
    pAA *= w; pBB *= w; pAB *= w;

    // wave32 butterfly reduction
#pragma unroll
    for (int off = 16; off >= 1; off >>= 1) {
        pAA += __shfl_xor(pAA, off, 32);
        pBB += __shfl_xor(pBB, off, 32);
        pAB += __shfl_xor(pAB, off, 32);
    }

    if (lane == 0) {
        atomicAdd(&acc[batch * 3 + 0], pAA);
        atomicAdd(&acc[batch * 3 + 1], pBB);
        atomicAdd(&acc[batch * 3 + 2], pAB);
    }
}

__global__ void schatten_finalize(const float* __restrict__ acc,
                                  const float* __restrict__ labels,
                                  float* __restrict__ out) {
    __shared__ float sm[NBATCH];
    const int t = threadIdx.x;
    float term = 0.0f;
    if (t < NBATCH) {
        const float Saa = fmaxf(acc[t * 3 + 0], 0.0f);
        const float Sbb = fmaxf(acc[t * 3 + 1], 0.0f);
        const float Sab = fmaxf(acc[t * 3 + 2], 0.0f);
        const float cross = sqrtf(Sab);                       // ||A B^T||_F
        const float na    = sqrtf(sqrtf(Saa));                // sqrt(||A A^T||_F)
        const float nb    = sqrtf(sqrtf(Sbb));
        const float o     = cross / (na * nb);
        const float d     = o - labels[t];
        term = d * d;
    }
    sm[t] = term;
    __syncthreads();
#pragma unroll
    for (int s = NBATCH / 2; s > 0; s >>= 1) {
        if (t < s) sm[t] += sm[t + s];
        __syncthreads();
    }
    if (t == 0) out[0] = sm[0] / (float)NBATCH;
}

extern "C" void kernel_launch(void* const* d_in, const int* in_sizes, int n_in,
                              void* d_out, int out_size, void* d_ws, size_t ws_size,
                              hipStream_t stream) {
    const float* rep_a  = (const float*)d_in[0];
    const float* rep_b  = (const float*)d_in[1];
    const float* labels = (const float*)d_in[2];
    float* out = (float*)d_out;
    float* acc = (float*)d_ws;   // NBATCH * 3 floats

    schatten_zero_acc<<<1, NBATCH * 3, 0, stream>>>(acc);

    const int total_waves = NBATCH * NPAIRS;                 // 9984
    const int blocks      = (total_waves + WPB - 1) / WPB;   // 1248
    schatten_gram_tiles<<<blocks, BLOCK, 0, stream>>>(rep_a, rep_b, acc);

    schatten_finalize<<<1, NBATCH, 0, stream>>>(acc, labels, out);
}